// CA1Replace_66211215835793
// MI455X (gfx1250) — compile-verified
//
#include <hip/hip_runtime.h>
#include <stdint.h>

#define CA_WIDTH 2048
#define CA_WORDS 64   // uint32 words per row
#define CA_LANES 32
#define EXP_THREADS 256

// (s & a) | (~s & b)  -> single v_bfi_b32
__device__ __forceinline__ uint32_t bfi(uint32_t s, uint32_t a, uint32_t b) {
    return (s & a) | (~s & b);
}

// ({hi,lo} >> sh)[31:0] -> v_alignbit_b32
__device__ __forceinline__ uint32_t alignbit(uint32_t hi, uint32_t lo, uint32_t sh) {
    return __builtin_amdgcn_alignbit(hi, lo, sh);
}

// Generic rule evaluation: idx = L + 2*C + 4*R, new = lookup[idx], bit-parallel.
// m[k] = all-ones if lookup[k]&1 else 0. Three v_bfi levels; "late" operand last.
__device__ __forceinline__ uint32_t rule_L_late(uint32_t L, uint32_t C, uint32_t R,
                                                const uint32_t m[8]) {
    uint32_t v00 = bfi(C, m[2], m[0]);
    uint32_t v01 = bfi(C, m[6], m[4]);
    uint32_t A0  = bfi(R, v01, v00);     // L=0 plane
    uint32_t v10 = bfi(C, m[3], m[1]);
    uint32_t v11 = bfi(C, m[7], m[5]);
    uint32_t A1  = bfi(R, v11, v10);     // L=1 plane
    return bfi(L, A1, A0);
}
__device__ __forceinline__ uint32_t rule_R_late(uint32_t L, uint32_t C, uint32_t R,
                                                const uint32_t m[8]) {
    uint32_t w00 = bfi(C, m[2], m[0]);
    uint32_t w10 = bfi(C, m[3], m[1]);
    uint32_t B0  = bfi(L, w10, w00);     // R=0 plane
    uint32_t w01 = bfi(C, m[6], m[4]);
    uint32_t w11 = bfi(C, m[7], m[5]);
    uint32_t B1  = bfi(L, w11, w01);     // R=1 plane
    return bfi(R, B1, B0);
}

// ---------------------------------------------------------------------------
// Phase 1: one wave32 per batch row, bit-packed evolution, history -> d_ws.
// Lane owns words (2*lane, 2*lane+1).
//   Fast path (lookup == rule 30): ghost word on each side of the 64-bit core;
//     ghosts stay valid for 32 local iterations (1 bit/iter decay), so the
//     cross-lane shuffle happens only every 32 iterations -> loop is pure-VALU
//     issue-bound instead of ds_bpermute latency-bound.
//   Generic path: per-iteration boundary shuffles + 3-level v_bfi mux.
// ---------------------------------------------------------------------------
__global__ __launch_bounds__(CA_LANES)
void ca_evolve_packed(const float* __restrict__ x,
                      const int* __restrict__ lookup,
                      uint32_t* __restrict__ hist,   // [B][iters+1][CA_WORDS]
                      int iters) {
    const int b    = blockIdx.x;
    const int lane = threadIdx.x & (CA_LANES - 1);

    uint32_t m[8];
    uint32_t rule8 = 0u;
#pragma unroll
    for (int k = 0; k < 8; ++k) {
        uint32_t bit = (uint32_t)(lookup[k] & 1);
        m[k] = 0u - bit;
        rule8 |= bit << k;
    }

    // --- pack initial state via coalesced loads + ballot ---
    const float* row = x + (size_t)b * CA_WIDTH;
    uint32_t c0 = 0u, c1 = 0u;
    for (int w = 0; w < CA_WORDS; ++w) {
        unsigned long long bal = __ballot(row[w * 32 + lane] >= 0.5f);
        uint32_t wv = (uint32_t)bal;      // wave32: low 32 bits
        if (w == 2 * lane)     c0 = wv;
        if (w == 2 * lane + 1) c1 = wv;
    }

    uint32_t* out = hist + (size_t)b * (size_t)(iters + 1) * CA_WORDS;
    *(uint2*)(out + 2 * lane) = make_uint2(c0, c1);   // t = 0
    out += CA_WORDS;

    const int lm = (lane + CA_LANES - 1) & (CA_LANES - 1);
    const int lp = (lane + 1) & (CA_LANES - 1);

    if (rule8 == 30u) {
        // idx = L + 2C + 4R with RULE=30  <=>  new = R ^ (C | L)
        // (verified bit-by-bit against the lookup table)
        uint32_t pg = 0u, ng = 0u;  // left/right ghost words
        for (int t = 0; t < iters; ++t) {
            if ((t & 31) == 0) {    // refresh ghosts: exact neighbor core words
                pg = (uint32_t)__shfl((int)c1, lm, CA_LANES);
                ng = (uint32_t)__shfl((int)c0, lp, CA_LANES);
            }
            // left ghost: bit0 garbage creeps up 1 bit/iter (never reaches
            // bit31 within 32 iters, which is all the core consumes)
            uint32_t Lp = pg << 1;                 // bit0 invalid, ok
            uint32_t Rp = alignbit(c0, pg, 1);
            uint32_t np = Rp ^ (pg | Lp);
            // core word 0
            uint32_t La = alignbit(c0, pg, 31);
            uint32_t Ra = alignbit(c1, c0, 1);
            uint32_t na = Ra ^ (c0 | La);
            // core word 1
            uint32_t Lb = alignbit(c1, c0, 31);
            uint32_t Rb = alignbit(ng, c1, 1);
            uint32_t nb = Rb ^ (c1 | Lb);
            // right ghost: bit31 garbage creeps down 1 bit/iter
            uint32_t Ln = alignbit(ng, c1, 31);
            uint32_t Rn = ng >> 1;                 // bit31 invalid, ok
            uint32_t nn = Rn ^ (ng | Ln);

            pg = np; c0 = na; c1 = nb; ng = nn;
            *(uint2*)(out + 2 * lane) = make_uint2(c0, c1);
            out += CA_WORDS;
        }
    } else {
        for (int t = 0; t < iters; ++t) {
            // boundary words from neighbor lanes (wrap across lane 0/31)
            uint32_t pw = (uint32_t)__shfl((int)c1, lm, CA_LANES);
            uint32_t nw = (uint32_t)__shfl((int)c0, lp, CA_LANES);

            uint32_t L0 = alignbit(c0, pw, 31);
            uint32_t R0 = alignbit(c1, c0, 1);
            uint32_t L1 = alignbit(c1, c0, 31);
            uint32_t R1 = alignbit(nw, c1, 1);

            uint32_t n0 = rule_L_late(L0, c0, R0, m);
            uint32_t n1 = rule_R_late(L1, c1, R1, m);
            c0 = n0; c1 = n1;

            *(uint2*)(out + 2 * lane) = make_uint2(c0, c1);
            out += CA_WORDS;
        }
    }
}

// ---------------------------------------------------------------------------
// Phase 2: packed bits -> floats, pure streaming (HBM-bound, ~134 MB out).
// Packed words are staged global->LDS via the CDNA5 async DMA path
// (GLOBAL_LOAD_ASYNC_TO_LDS_B32, ASYNCcnt, s_wait_asynccnt), then expanded.
// ---------------------------------------------------------------------------
__global__ __launch_bounds__(EXP_THREADS)
void ca_expand(const uint32_t* __restrict__ hist,
               float* __restrict__ out,
               long long nwords) {
    __shared__ uint32_t smem[EXP_THREADS];
    long long g = (long long)blockIdx.x * blockDim.x + threadIdx.x;
    if (g >= nwords) return;

    // 32-bit LDS offset for the async instruction's VDST operand
    // (generic LDS addresses carry the LDS offset in addr[31:0], ISA 10.2)
    uint32_t lds_addr = (uint32_t)(size_t)(void*)&smem[threadIdx.x];
    uint64_t gaddr    = (uint64_t)(size_t)(hist + g);

    asm volatile("global_load_async_to_lds_b32 %0, %1, off"
                 :
                 : "v"(lds_addr), "v"(gaddr)
                 : "memory");
    asm volatile("s_wait_asynccnt 0x0" ::: "memory");

    uint32_t w = smem[threadIdx.x];
    float4* o = (float4*)(out + g * 32);
#pragma unroll
    for (int j = 0; j < 8; ++j) {
        float4 v;
        v.x = (w & (1u << (4 * j + 0))) ? 1.0f : 0.0f;
        v.y = (w & (1u << (4 * j + 1))) ? 1.0f : 0.0f;
        v.z = (w & (1u << (4 * j + 2))) ? 1.0f : 0.0f;
        v.w = (w & (1u << (4 * j + 3))) ? 1.0f : 0.0f;
        o[j] = v;
    }
}

// ---------------------------------------------------------------------------
// Fallback (only if ws is too small for packed history): evolve and write
// floats directly. Correct but slower; deterministic since ws_size is fixed.
// ---------------------------------------------------------------------------
__device__ __forceinline__ void store_word_floats(float* dst, uint32_t w) {
#pragma unroll
    for (int j = 0; j < 8; ++j) {
        float4 v;
        v.x = (w & (1u << (4 * j + 0))) ? 1.0f : 0.0f;
        v.y = (w & (1u << (4 * j + 1))) ? 1.0f : 0.0f;
        v.z = (w & (1u << (4 * j + 2))) ? 1.0f : 0.0f;
        v.w = (w & (1u << (4 * j + 3))) ? 1.0f : 0.0f;
        ((float4*)dst)[j] = v;
    }
}

__global__ __launch_bounds__(CA_LANES)
void ca_evolve_direct(const float* __restrict__ x,
                      const int* __restrict__ lookup,
                      float* __restrict__ out,    // [B][iters+1][CA_WIDTH]
                      int iters) {
    const int b    = blockIdx.x;
    const int lane = threadIdx.x & (CA_LANES - 1);

    uint32_t m[8];
#pragma unroll
    for (int k = 0; k < 8; ++k) m[k] = 0u - (uint32_t)(lookup[k] & 1);

    const float* row = x + (size_t)b * CA_WIDTH;
    uint32_t c0 = 0u, c1 = 0u;
    for (int w = 0; w < CA_WORDS; ++w) {
        unsigned long long bal = __ballot(row[w * 32 + lane] >= 0.5f);
        uint32_t wv = (uint32_t)bal;
        if (w == 2 * lane)     c0 = wv;
        if (w == 2 * lane + 1) c1 = wv;
    }

    float* orow = out + (size_t)b * (size_t)(iters + 1) * CA_WIDTH;
    store_word_floats(orow + lane * 64, c0);
    store_word_floats(orow + lane * 64 + 32, c1);
    orow += CA_WIDTH;

    const int lm = (lane + CA_LANES - 1) & (CA_LANES - 1);
    const int lp = (lane + 1) & (CA_LANES - 1);

    for (int t = 0; t < iters; ++t) {
        uint32_t pw = (uint32_t)__shfl((int)c1, lm, CA_LANES);
        uint32_t nw = (uint32_t)__shfl((int)c0, lp, CA_LANES);
        uint32_t L0 = alignbit(c0, pw, 31);
        uint32_t R0 = alignbit(c1, c0, 1);
        uint32_t L1 = alignbit(c1, c0, 31);
        uint32_t R1 = alignbit(nw, c1, 1);
        uint32_t n0 = rule_L_late(L0, c0, R0, m);
        uint32_t n1 = rule_R_late(L1, c1, R1, m);
        c0 = n0; c1 = n1;
        store_word_floats(orow + lane * 64, c0);
        store_word_floats(orow + lane * 64 + 32, c1);
        orow += CA_WIDTH;
    }
}

extern "C" void kernel_launch(void* const* d_in, const int* in_sizes, int n_in,
                              void* d_out, int out_size, void* d_ws, size_t ws_size,
                              hipStream_t stream) {
    const float* x      = (const float*)d_in[0];   // [B, 2048] float32
    const int*   lookup = (const int*)d_in[1];     // [8] int
    float*       out    = (float*)d_out;           // [B, iters+1, 2048] float32

    const int B     = in_sizes[0] / CA_WIDTH;         // 16
    const int iters = out_size / (B * CA_WIDTH) - 1;  // 1024

    const size_t need = (size_t)B * (size_t)(iters + 1) * CA_WORDS * sizeof(uint32_t);

    if (ws_size >= need) {
        // Phase 1: bit-packed evolution into workspace (one wave32 per row)
        ca_evolve_packed<<<B, CA_LANES, 0, stream>>>(x, lookup, (uint32_t*)d_ws, iters);
        // Phase 2: async-LDS staged expansion, bits -> floats
        long long nwords = (long long)out_size / 32;
        long long blocks = (nwords + EXP_THREADS - 1) / EXP_THREADS;
        ca_expand<<<dim3((unsigned)blocks), EXP_THREADS, 0, stream>>>(
            (const uint32_t*)d_ws, out, nwords);
    } else {
        ca_evolve_direct<<<B, CA_LANES, 0, stream>>>(x, lookup, out, iters);
    }
}